// VQEMA_57037165691628
// MI455X (gfx1250) — compile-verified
//
#include <hip/hip_runtime.h>
#include <hip/hip_bf16.h>

typedef __attribute__((ext_vector_type(2))) float v2f;
typedef __attribute__((ext_vector_type(8))) float v8f;

#define NUM_EMBED   4096
#define EMBED_DIM   16
#define DECAYF      0.99f
#define OMDECAYF    0.01f
#define EPSF        1e-5f

#define N_TOKENS    65536    // 16 * 64 * 64
#define HW          4096     // 64*64
#define Z_NUMEL     1048576  // 16*16*64*64

// ---------------- workspace layout (float units) ----------------
// [0,      4096)   n histogram
// [4096,  69632)   embed_sum (4096*16)
// [69632]          loss accumulator
// [69633]          n_total accumulator
// [69634, 73730)   enorm  (||e_k||^2)
// [73730,139266)   int32 indices (reinterpreted)
#define WS_N       0
#define WS_ESUM    4096
#define WS_LOSS    69632
#define WS_NTOTAL  69633
#define WS_ENORM   69634
#define WS_IDX     73730
#define WS_ZERO_CNT 69634   // floats to zero each launch

// ---------------- output layout (float units) ----------------
#define O_ZQ    0
#define O_LOSS  1048576
#define O_IDX   1048577
#define O_NCS   1114113
#define O_NEA   1118209
#define O_NW    1183745

__global__ __launch_bounds__(256) void vq_zero_kernel(float* __restrict__ ws) {
    int i = blockIdx.x * blockDim.x + threadIdx.x;
    if (i < WS_ZERO_CNT) ws[i] = 0.0f;
}

__global__ __launch_bounds__(256) void vq_enorm_kernel(const float* __restrict__ E,
                                                       float* __restrict__ enorm) {
    int k = blockIdx.x * blockDim.x + threadIdx.x;
    if (k < NUM_EMBED) {
        float s = 0.0f;
        #pragma unroll
        for (int c = 0; c < EMBED_DIM; ++c) { float v = E[k * EMBED_DIM + c]; s = fmaf(v, v, s); }
        enorm[k] = s;
    }
}

// One wave = 16 tokens vs all 4096 codes.  score = ||e||^2 - 2*(z.e); argmin.
__global__ __launch_bounds__(256) void vq_argmin_kernel(const float* __restrict__ z,
                                                        const float* __restrict__ E,
                                                        const float* __restrict__ enorm,
                                                        float* __restrict__ idx_f,
                                                        int*   __restrict__ idx_i) {
    const int lane      = threadIdx.x & 31;
    const int wave      = threadIdx.x >> 5;
    const int tokenBase = blockIdx.x * 128 + wave * 16;

    const int m    = lane & 15;          // A row / token within block of 16
    const int koff = (lane >> 4) << 1;   // 0 for lanes 0-15, 2 for lanes 16-31

    // A matrix: token tok, features strided by HW in [B,C,H,W]
    const int tok = tokenBase + m;
    const int b   = tok >> 12;
    const int hw  = tok & (HW - 1);
    const float* zp = z + b * (EMBED_DIM * HW) + hw;

    v2f a[4];
    #pragma unroll
    for (int q = 0; q < 4; ++q) {
        const int c0 = 4 * q + koff;
        a[q].x = zp[c0 * HW];
        a[q].y = zp[(c0 + 1) * HW];
    }

    float bestv[8];
    int   besti[8];
    #pragma unroll
    for (int r = 0; r < 8; ++r) { bestv[r] = 3.4e38f; besti[r] = 0; }

    const int jn = lane & 15;            // B column within tile
    for (int t = 0; t < NUM_EMBED / 16; ++t) {
        const int j = t * 16 + jn;
        const float* ep = E + j * EMBED_DIM;
        v2f bb[4];
        #pragma unroll
        for (int q = 0; q < 4; ++q) {
            const int c0 = 4 * q + koff;
            bb[q].x = ep[c0];
            bb[q].y = ep[c0 + 1];
        }
        v8f acc = {};
        #pragma unroll
        for (int q = 0; q < 4; ++q) {
            acc = __builtin_amdgcn_wmma_f32_16x16x4_f32(
                /*neg_a=*/false, a[q], /*neg_b=*/false, bb[q],
                /*c_mod=*/(short)0, acc, /*reuse_a=*/false, /*reuse_b=*/false);
        }
        const float en = enorm[j];
        #pragma unroll
        for (int r = 0; r < 8; ++r) {
            const float s = fmaf(-2.0f, acc[r], en);  // ||e||^2 - 2 z.e
            if (s < bestv[r]) { bestv[r] = s; besti[r] = j; }
        }
    }

    // min-reduce across the 16 lanes holding the same token row (xor<16 stays in half)
    #pragma unroll
    for (int off = 1; off < 16; off <<= 1) {
        #pragma unroll
        for (int r = 0; r < 8; ++r) {
            const float ov = __shfl_xor(bestv[r], off, 32);
            const int   oi = __shfl_xor(besti[r], off, 32);
            if (ov < bestv[r] || (ov == bestv[r] && oi < besti[r])) {
                bestv[r] = ov; besti[r] = oi;
            }
        }
    }

    if ((lane & 15) == 0) {
        const int hi = lane >> 4;        // lane0 -> tokens +0..7, lane16 -> +8..15
        #pragma unroll
        for (int r = 0; r < 8; ++r) {
            const int tk = tokenBase + r + 8 * hi;
            idx_i[tk] = besti[r];
            idx_f[tk] = (float)besti[r];
        }
    }
}

// One thread per token: write z_q (= z_q_st forward value), accumulate loss,
// histogram n, and segment-sum of z_flat.
__global__ __launch_bounds__(256) void vq_assign_kernel(const float* __restrict__ z,
                                                        const float* __restrict__ E,
                                                        const int*   __restrict__ idx_i,
                                                        float* __restrict__ zq_out,
                                                        float* __restrict__ ws_n,
                                                        float* __restrict__ ws_esum,
                                                        float* __restrict__ ws_loss) {
    const int tok = blockIdx.x * blockDim.x + threadIdx.x;
    const int k   = idx_i[tok];
    const float* ep = E + k * EMBED_DIM;
    const int b  = tok >> 12;
    const int hw = tok & (HW - 1);
    const float* zp = z + b * (EMBED_DIM * HW) + hw;
    float*       op = zq_out + b * (EMBED_DIM * HW) + hw;

    float err = 0.0f;
    #pragma unroll
    for (int c = 0; c < EMBED_DIM; ++c) {
        const float e  = ep[c];
        const float zv = zp[c * HW];
        op[c * HW] = e;
        const float d = zv - e;
        err = fmaf(d, d, err);
        atomicAdd(&ws_esum[k * EMBED_DIM + c], zv);
    }
    atomicAdd(&ws_n[k], 1.0f);

    #pragma unroll
    for (int off = 16; off > 0; off >>= 1) err += __shfl_down(err, off, 32);
    if ((threadIdx.x & 31) == 0) atomicAdd(ws_loss, err);
}

__global__ __launch_bounds__(256) void vq_ema_kernel(const float* __restrict__ cluster_size,
                                                     const float* __restrict__ embed_avg,
                                                     const float* __restrict__ ws_n,
                                                     const float* __restrict__ ws_esum,
                                                     float* __restrict__ ncs_out,
                                                     float* __restrict__ nea_out,
                                                     float* __restrict__ ws_ntotal) {
    const int i = blockIdx.x * blockDim.x + threadIdx.x;
    if (i < NUM_EMBED) {
        const float ncs = cluster_size[i] * DECAYF + OMDECAYF * ws_n[i];
        ncs_out[i] = ncs;
        float s = ncs;
        #pragma unroll
        for (int off = 16; off > 0; off >>= 1) s += __shfl_down(s, off, 32);
        if ((threadIdx.x & 31) == 0) atomicAdd(ws_ntotal, s);
    }
    if (i < NUM_EMBED * EMBED_DIM) {
        nea_out[i] = embed_avg[i] * DECAYF + OMDECAYF * ws_esum[i];
    }
}

__global__ __launch_bounds__(256) void vq_final_kernel(const float* __restrict__ ncs_out,
                                                       const float* __restrict__ nea_out,
                                                       const float* __restrict__ ws_ntotal,
                                                       const float* __restrict__ ws_loss,
                                                       float* __restrict__ nw_out,
                                                       float* __restrict__ loss_out) {
    const int i = blockIdx.x * blockDim.x + threadIdx.x;
    if (i == 0) loss_out[0] = 1.25f * ws_loss[0] / (float)Z_NUMEL;
    if (i < NUM_EMBED * EMBED_DIM) {
        const int   k  = i >> 4;
        const float nt = ws_ntotal[0];
        const float cs = (ncs_out[k] + EPSF) / (nt + (float)NUM_EMBED * EPSF) * nt;
        nw_out[i] = nea_out[i] / cs;
    }
}

extern "C" void kernel_launch(void* const* d_in, const int* in_sizes, int n_in,
                              void* d_out, int out_size, void* d_ws, size_t ws_size,
                              hipStream_t stream) {
    const float* z            = (const float*)d_in[0];
    const float* E            = (const float*)d_in[1];
    const float* cluster_size = (const float*)d_in[2];
    const float* embed_avg    = (const float*)d_in[3];

    float* out     = (float*)d_out;
    float* o_zq    = out + O_ZQ;
    float* o_loss  = out + O_LOSS;
    float* o_idx   = out + O_IDX;
    float* o_ncs   = out + O_NCS;
    float* o_nea   = out + O_NEA;
    float* o_nw    = out + O_NW;

    float* ws        = (float*)d_ws;
    float* ws_n      = ws + WS_N;
    float* ws_esum   = ws + WS_ESUM;
    float* ws_loss   = ws + WS_LOSS;
    float* ws_ntotal = ws + WS_NTOTAL;
    float* ws_enorm  = ws + WS_ENORM;
    int*   ws_idx    = (int*)(ws + WS_IDX);

    vq_zero_kernel<<<(WS_ZERO_CNT + 255) / 256, 256, 0, stream>>>(ws);
    vq_enorm_kernel<<<NUM_EMBED / 256, 256, 0, stream>>>(E, ws_enorm);
    vq_argmin_kernel<<<N_TOKENS / 128, 256, 0, stream>>>(z, E, ws_enorm, o_idx, ws_idx);
    vq_assign_kernel<<<N_TOKENS / 256, 256, 0, stream>>>(z, E, ws_idx, o_zq, ws_n, ws_esum, ws_loss);
    vq_ema_kernel<<<(NUM_EMBED * EMBED_DIM) / 256, 256, 0, stream>>>(cluster_size, embed_avg,
                                                                     ws_n, ws_esum, o_ncs, o_nea, ws_ntotal);
    vq_final_kernel<<<(NUM_EMBED * EMBED_DIM) / 256, 256, 0, stream>>>(o_ncs, o_nea, ws_ntotal,
                                                                       ws_loss, o_nw, o_loss);
}